// AddAttention_56444460204424
// MI455X (gfx1250) — compile-verified
//
#include <hip/hip_runtime.h>
#include <hip/hip_bf16.h>

// ---------------------------------------------------------------------------
// Additive (Bahdanau) attention scores for MI455X (gfx1250).
//   pq = q @ Wq^T   (B,Q,H)    -- f32 WMMA 16x16x4
//   pk = k @ Wk^T   (B,K,H)    -- f32 WMMA 16x16x4
//   scores[b,q,k] = sum_h w[h] * tanh(pq[b,q,h] + pk[b,k,h]) ; mask k>=S[b]
// Dominant cost: 268M hardware tanh + FMA (trans-VALU bound, mem floor ~1.3us)
// ---------------------------------------------------------------------------

#define B_  4
#define Q_  512
#define K_  512
#define D_  512
#define H_  256

typedef __attribute__((ext_vector_type(2))) float v2f;
typedef __attribute__((ext_vector_type(8))) float v8f;

__device__ __forceinline__ float fast_tanh(float x) {
#if __has_builtin(__builtin_amdgcn_tanhf)
  return __builtin_amdgcn_tanhf(x);
#else
  float r;
  asm volatile("v_tanh_f32 %0, %1" : "=v"(r) : "v"(x));
  return r;
#endif
}

// ---------------------------------------------------------------------------
// Projection GEMM: dst[b,r,h] = sum_d src[b,r,d] * W[h,d]
// src:(B,R,D) W:(H,D) dst:(B,R,H).  One wave per 16x16 tile, 4 waves/block.
// grid = (R/16, H/64, B), block = 128.
// A 16x4 f32 layout: lane<16 -> {A[M=l][K=0],A[M=l][K=1]}, lane>=16 -> K=2,3.
// B 4x16  f32 layout: lane<16 -> {B[K=0][N=l],B[K=1][N=l]}, lane>=16 -> K=2,3.
// ---------------------------------------------------------------------------
__global__ __launch_bounds__(128)
void proj_wmma(const float* __restrict__ src, const float* __restrict__ W,
               float* __restrict__ dst, int R, int D, int H) {
  const int lane  = threadIdx.x & 31;
  const int wave  = threadIdx.x >> 5;
  const int r0    = blockIdx.x * 16;
  const int h0    = (blockIdx.y * 4 + wave) * 16;
  const int b     = blockIdx.z;

  const int half = lane >> 4;   // 0: K=0,1   1: K=2,3
  const int l16  = lane & 15;

  const float* aPtr = src + ((size_t)b * R + (r0 + l16)) * D + 2 * half;
  const float* bPtr = W + (size_t)(h0 + l16) * D + 2 * half;

  v8f acc = {};
  for (int d0 = 0; d0 < D; d0 += 4) {
    v2f a  = *(const v2f*)(aPtr + d0);
    v2f bb = *(const v2f*)(bPtr + d0);
    // D = A(16x4) * B(4x16) + C ; emits v_wmma_f32_16x16x4_f32
    acc = __builtin_amdgcn_wmma_f32_16x16x4_f32(
        /*neg_a=*/false, a, /*neg_b=*/false, bb,
        /*c_mod=*/(short)0, acc, /*reuse_a=*/false, /*reuse_b=*/false);
  }

  // C/D layout: VGPR g: lanes 0-15 -> M=g, N=lane; lanes 16-31 -> M=g+8.
  float* dstB = dst + (size_t)b * R * H;
  const int col     = h0 + l16;
  const int rowBase = r0 + half * 8;
#pragma unroll
  for (int g = 0; g < 8; ++g) {
    dstB[(size_t)(rowBase + g) * H + col] = acc[g];
  }
}

// ---------------------------------------------------------------------------
// Score kernel: 16(q) x 32(k) tile per block, 512 threads (16 waves).
// LDS: pq tile 16x256 (16KB, broadcast reads), pk tile 32x(256+1) padded
// stride 257 dwords == 1 mod 64 banks -> conflict-free per-lane reads, w 1KB.
// Inner loop per element: ds_load + ds_load + v_add + v_tanh + v_fmac.
// ---------------------------------------------------------------------------
#define TQ 16
#define TK 32
#define PKS (H_ + 1)   // 257, bank-conflict-free stride

__global__ __launch_bounds__(512)
void scores_kernel(const float* __restrict__ pq, const float* __restrict__ pk,
                   const float* __restrict__ w, const int* __restrict__ S,
                   float* __restrict__ out) {
  __shared__ float s_pq[TQ * H_];    // 16 KB
  __shared__ float s_pk[TK * PKS];   // ~32 KB
  __shared__ float s_w[H_];          // 1 KB

  const int tid = threadIdx.x;       // 0..511
  const int b   = blockIdx.z;
  const int q0  = blockIdx.y * TQ;
  const int k0  = blockIdx.x * TK;

  const float* pqB = pq + ((size_t)b * Q_ + q0) * H_;  // 16x256 contiguous
  const float* pkB = pk + ((size_t)b * K_ + k0) * H_;  // 32x256 contiguous

  // pq tile: 4096 floats, float4-vectorized, coalesced.
  {
    const float4* s4 = (const float4*)pqB;
    float4*       d4 = (float4*)s_pq;
#pragma unroll
    for (int i = 0; i < (TQ * H_ / 4) / 512; ++i)
      d4[tid + i * 512] = s4[tid + i * 512];
  }
  // pk tile: 8192 floats into padded LDS rows (scalar stores, coalesced gmem).
  for (int j = tid; j < TK * H_; j += 512) {
    const int row = j >> 8;          // /256
    const int col = j & (H_ - 1);
    s_pk[row * PKS + col] = pkB[j];
  }
  if (tid < H_) s_w[tid] = w[tid];
  __syncthreads();

  const int x  = tid & 31;           // k within tile (lane id -> coalesced out)
  const int y  = tid >> 5;           // q within tile (wave-uniform)
  const int kg = k0 + x;
  const int Sb = S[b];

  const float* pqRow = s_pq + y * H_;    // wave-uniform address -> broadcast
  const float* pkRow = s_pk + x * PKS;   // stride 257 dwords -> no conflicts

  float acc = 0.f;
#pragma unroll 4
  for (int h = 0; h < H_; h += 4) {
    const float4 pq4 = *(const float4*)(pqRow + h);   // broadcast b128
    const float4 w4  = *(const float4*)(s_w + h);     // broadcast b128
    acc += w4.x * fast_tanh(pq4.x + pkRow[h + 0]);
    acc += w4.y * fast_tanh(pq4.y + pkRow[h + 1]);
    acc += w4.z * fast_tanh(pq4.z + pkRow[h + 2]);
    acc += w4.w * fast_tanh(pq4.w + pkRow[h + 3]);
  }

  const float result = (kg >= Sb) ? -__builtin_inff() : acc;
  out[((size_t)b * Q_ + (q0 + y)) * K_ + kg] = result;   // coalesced per wave
}

// ---------------------------------------------------------------------------
extern "C" void kernel_launch(void* const* d_in, const int* in_sizes, int n_in,
                              void* d_out, int out_size, void* d_ws, size_t ws_size,
                              hipStream_t stream) {
  const float* q  = (const float*)d_in[0];
  const float* k  = (const float*)d_in[1];
  // d_in[2] = v : unused by the reference output
  const int*   S  = (const int*)d_in[3];
  const float* Wq = (const float*)d_in[4];
  const float* Wk = (const float*)d_in[5];
  const float* w  = (const float*)d_in[6];
  float* out = (float*)d_out;

  // Workspace: pq (2 MB) | pk (2 MB)
  float* pq = (float*)d_ws;
  float* pk = pq + (size_t)B_ * Q_ * H_;

  dim3 gq(Q_ / 16, H_ / 64, B_);
  proj_wmma<<<gq, 128, 0, stream>>>(q, Wq, pq, Q_, D_, H_);
  dim3 gk(K_ / 16, H_ / 64, B_);
  proj_wmma<<<gk, 128, 0, stream>>>(k, Wk, pk, K_, D_, H_);

  dim3 gs(K_ / TK, Q_ / TQ, B_);
  scores_kernel<<<gs, TQ * TK, 0, stream>>>(pq, pk, w, S, out);
}